// GeneralConvNd_27032524161782
// MI455X (gfx1250) — compile-verified
//
#include <hip/hip_runtime.h>

typedef float v2f __attribute__((ext_vector_type(2)));
typedef float v8f __attribute__((ext_vector_type(8)));

#define KSZ   7
#define PADD  3
#define NCOUT 64
#define HH    384
#define WW    384
#define NB    8

#define WT    16            // pixels along W per wave tile (WMMA M)
#define HT    8             // rows per sub-tile (one row per wave, 8 waves)
#define SUBT  6             // sub-tiles per workgroup
#define ROWS_PER_WG (HT * SUBT)   // 48

#define KPAD  52            // 49 rounded up to multiple of 4
#define WROW  80            // Wlds row stride (dwords): lane halves hit disjoint banks
#define XW    (WT + KSZ - 1)      // 22
#define XH    (HT + KSZ - 1)      // 14

__global__ __launch_bounds__(256)
void conv7x7_wmma_f32(const float* __restrict__ x,
                      const float* __restrict__ Wg,
                      const float* __restrict__ bias,
                      float* __restrict__ out)
{
    __shared__ float Wlds[KPAD * WROW];   // 16.6 KB, K-padded with zeros
    __shared__ float xlds[XH * XW];       // 1.2 KB halo tile

    const int tid  = threadIdx.x;
    const int lane = tid & 31;
    const int wave = tid >> 5;
    const int half = lane >> 4;   // lane half selects K offset (+2)
    const int nl   = lane & 15;

    const int w0    = blockIdx.x * WT;
    const int hbase = blockIdx.y * ROWS_PER_WG;
    const int bb    = blockIdx.z;

    // ---- stage weights into LDS, rows 49..51 zeroed ----
    for (int i = tid; i < KPAD * NCOUT; i += 256) {
        int k = i >> 6, n = i & 63;
        Wlds[k * WROW + n] = (k < KSZ * KSZ) ? Wg[k * NCOUT + n] : 0.0f;
    }
    __syncthreads();

    // ---- hoist all B fragments to registers (shared by every pixel tile) ----
    // B 4x16 f32 layout: lane = N col; lanes 0-15 rows {k0,k0+1}, lanes 16-31 rows {k0+2,k0+3}
    v2f bfrag[13][4];
#pragma unroll
    for (int s = 0; s < 13; ++s) {
        const int k = 4 * s + 2 * half;          // 0..50 (padded rows are zero)
#pragma unroll
        for (int t = 0; t < 4; ++t) {
            bfrag[s][t].x = Wlds[(k    ) * WROW + t * 16 + nl];
            bfrag[s][t].y = Wlds[(k + 1) * WROW + t * 16 + nl];
        }
    }

    // ---- per-lane A-fragment LDS offsets (row term added per sub-tile) ----
    int aoff[13][2];
#pragma unroll
    for (int s = 0; s < 13; ++s) {
#pragma unroll
        for (int j = 0; j < 2; ++j) {
            int k = 4 * s + 2 * half + j;
            k = (k > 48) ? 48 : k;               // k>=49 pairs with zero B rows
            const int kh = k / KSZ;
            const int kw = k - KSZ * kh;
            aoff[s][j] = kh * XW + kw + nl;      // nl = pixel index M within tile
        }
    }

    float bv[4];
#pragma unroll
    for (int t = 0; t < 4; ++t) bv[t] = bias[t * 16 + nl];

    for (int st = 0; st < SUBT; ++st) {
        const int h0 = hbase + st * HT;

        __syncthreads();                          // xlds reuse across sub-tiles
        // ---- halo load with zero padding ----
        for (int i = tid; i < XH * XW; i += 256) {
            const int r  = i / XW;
            const int c  = i - r * XW;
            const int gh = h0 - PADD + r;
            const int gw = w0 - PADD + c;
            float v = 0.0f;
            if (gh >= 0 && gh < HH && gw >= 0 && gw < WW)
                v = x[((size_t)bb * HH + gh) * WW + gw];
            xlds[i] = v;
        }
        __syncthreads();

        const int abase = wave * XW;              // this wave's pixel row

        v8f acc0 = {}, acc1 = {}, acc2 = {}, acc3 = {};
#pragma unroll
        for (int s = 0; s < 13; ++s) {
            v2f a;
            a.x = xlds[abase + aoff[s][0]];
            a.y = xlds[abase + aoff[s][1]];
            acc0 = __builtin_amdgcn_wmma_f32_16x16x4_f32(false, a, false, bfrag[s][0], (short)0, acc0, false, false);
            acc1 = __builtin_amdgcn_wmma_f32_16x16x4_f32(false, a, false, bfrag[s][1], (short)0, acc1, false, false);
            acc2 = __builtin_amdgcn_wmma_f32_16x16x4_f32(false, a, false, bfrag[s][2], (short)0, acc2, false, false);
            acc3 = __builtin_amdgcn_wmma_f32_16x16x4_f32(false, a, false, bfrag[s][3], (short)0, acc3, false, false);
        }

        // ---- store: VGPR r -> pixel m = r + 8*half, col n = t*16 + nl ----
        float* op = out + (((size_t)bb * HH + (h0 + wave)) * WW + w0) * NCOUT;
#pragma unroll
        for (int r = 0; r < 8; ++r) {
            const int m = r + 8 * half;
            float* p = op + m * NCOUT + nl;
            p[0]  = acc0[r] + bv[0];
            p[16] = acc1[r] + bv[1];
            p[32] = acc2[r] + bv[2];
            p[48] = acc3[r] + bv[3];
        }
    }
}

extern "C" void kernel_launch(void* const* d_in, const int* in_sizes, int n_in,
                              void* d_out, int out_size, void* d_ws, size_t ws_size,
                              hipStream_t stream) {
    const float* x  = (const float*)d_in[0];   // [8,384,384]
    const float* Wg = (const float*)d_in[1];   // [49,64]
    const float* b  = (const float*)d_in[2];   // [64]
    float* out = (float*)d_out;                // [8,384,384,64]

    dim3 grid(WW / WT, HH / ROWS_PER_WG, NB);  // 24 x 8 x 8 = 1536 WGs
    conv7x7_wmma_f32<<<grid, 256, 0, stream>>>(x, Wg, b, out);
}